// DWTTransform_1185410974328
// MI455X (gfx1250) — compile-verified
//
#include <hip/hip_runtime.h>

// Haar DWT2D (8,32,512,512) f32 -> (8,128,257,257) f32.
// Pure bandwidth problem (~0.52 GB total traffic -> ~23 us at 23.3 TB/s).
// The 4-tap x 4-subband dot product runs on V_WMMA_F32_16X16X4_F32
// (D = A(16 windows x 4 taps) * B(4 taps x 4 subbands), exact in f32).
// One wave per output row; loop over 32-column tile pairs so per-wave setup
// (division, reflect clamps, B load, pointers) is amortized and stores are
// 16B/lane global_store_b128 -> high bytes-per-instruction, which is what
// decides whether this kernel is HBM-limited instead of issue-limited.

typedef __attribute__((ext_vector_type(2))) float v2f;
typedef __attribute__((ext_vector_type(4))) float v4f;                 // 16B aligned (LDS)
typedef __attribute__((ext_vector_type(4), aligned(4))) float v4fu;    // 4B aligned (global out)
typedef __attribute__((ext_vector_type(8))) float v8f;

#define BB    8
#define CC    32
#define HH    512
#define WW    512
#define H2    257
#define W2    257
#define PLANE (257 * 257)
#define WPB   8   // waves per block

__global__ __launch_bounds__(256)
void haar_dwt2d_wmma(const float* __restrict__ x,
                     const float* __restrict__ filt,
                     float* __restrict__ out,
                     int n_rows) {
  const int lane = threadIdx.x & 31;
  const int wave = threadIdx.x >> 5;
  const int half = lane >> 4;     // 0: top window row (K0,K1); 1: bottom (K2,K3)
  const int l    = lane & 15;     // window index within a 16-wide tile

  __shared__ float sbuf[WPB][128];   // 32 cols x 4 subbands staged per wave

  const int unit = blockIdx.x * WPB + wave;   // wave-uniform row id (b,c,i)
  if (unit >= n_rows) return;

  int i   = unit % H2;
  int t2  = unit / H2;
  int cch = t2 % CC;
  int b   = t2 / CC;

  // Reflect-padded source rows for output row i.
  int r0 = 2 * i - 1; if (r0 < 0) r0 = 1;            // i==0   -> x[1]
  int r1 = 2 * i;     if (r1 > HH - 1) r1 = HH - 2;  // i==256 -> x[510]

  const float* plane_in = x + (size_t)(b * CC + cch) * (HH * WW);
  const float* rowp     = plane_in + (size_t)(half ? r1 : r0) * WW;

  // B matrix 4x16 f32, loaded once: B[k,n] = filt[n*4+k] (n<4), else 0.
  // lanes 0-15 hold K0/K1 in VGPR0/1 at N=lane; lanes 16-31 hold K2/K3.
  v2f bmat; bmat.x = 0.0f; bmat.y = 0.0f;
  if (l < 4) {
    const int k = half * 2;
    bmat.x = filt[l * 4 + k];
    bmat.y = filt[l * 4 + k + 1];
  }

  // Per-lane b128 store target: subband nsub, 4-column group q within a
  // 32-column tile pair.
  const int nsub = lane >> 3;
  const int q    = lane & 7;
  float* obase = out + (size_t)(b * 4 * CC + nsub * CC + cch) * PLANE
                     + (size_t)i * W2 + q * 4;

  // LDS addresses: writer stages D (lanes with l<4 hold subband n=l, windows
  // m = 8*half + g in acc[g]); reader pulls one float4 per lane.
  float*       swr = &sbuf[wave][l * 32 + 8 * half];
  const float* srd = &sbuf[wave][nsub * 32 + q * 4];

  for (int tp = 0; tp < 8; ++tp) {      // columns [tp*32, tp*32+32)
#pragma unroll
    for (int s = 0; s < 2; ++s) {
      const int cu = tp * 64 + s * 32 + 2 * l;   // = 2*j for window j
      int ca = cu - 1; if (ca < 1) ca = 1;       // reflect: j==0 -> col 1
      v2f a;                                     // A 16x4: VGPR0=K(0|2), VGPR1=K(1|3)
      a.x = rowp[ca];
      a.y = rowp[cu];
      v8f acc = {};
      acc = __builtin_amdgcn_wmma_f32_16x16x4_f32(
          /*neg_a=*/false, a, /*neg_b=*/false, bmat,
          /*c_mod=*/(short)0, acc, /*reuse_a=*/false, /*reuse_b=*/false);
      if (l < 4) {
        v4f lo = { acc[0], acc[1], acc[2], acc[3] };
        v4f hi = { acc[4], acc[5], acc[6], acc[7] };
        *(v4f*)(swr + s * 16)     = lo;   // m = 8*half + 0..3
        *(v4f*)(swr + s * 16 + 4) = hi;   // m = 8*half + 4..7
      }
    }
    // Same-wave DS ops are in-order; fences only stop compiler reordering.
    __builtin_amdgcn_wave_barrier();
    asm volatile("s_wait_dscnt 0" ::: "memory");
    v4f v = *(const v4f*)srd;
    *(v4fu*)(obase + tp * 32) = v;        // coalesced 16B/lane store
    __builtin_amdgcn_wave_barrier();
  }

  // Tail column j=256 (swapped reflect pair): 4 lanes, one subband each.
  if (lane < 4) {
    const float* rp0 = plane_in + (size_t)r0 * WW;
    const float* rp1 = plane_in + (size_t)r1 * WW;
    const float ta = rp0[WW - 1], tb = rp0[WW - 2];   // C0=511, C1=510
    const float tc = rp1[WW - 1], td = rp1[WW - 2];
    const float v = filt[lane * 4 + 0] * ta + filt[lane * 4 + 1] * tb
                  + filt[lane * 4 + 2] * tc + filt[lane * 4 + 3] * td;
    out[(size_t)(b * 4 * CC + lane * CC + cch) * PLANE
        + (size_t)i * W2 + (W2 - 1)] = v;
  }
}

extern "C" void kernel_launch(void* const* d_in, const int* in_sizes, int n_in,
                              void* d_out, int out_size, void* d_ws, size_t ws_size,
                              hipStream_t stream) {
  const float* x    = (const float*)d_in[0];   // (8,32,512,512) f32
  const float* filt = (const float*)d_in[1];   // (4,2,2) f32 row-major
  float* out        = (float*)d_out;           // (8,128,257,257) f32

  const int n_rows = BB * CC * H2;             // one wave per output row
  const int blocks = (n_rows + WPB - 1) / WPB;
  haar_dwt2d_wmma<<<blocks, 256, 0, stream>>>(x, filt, out, n_rows);
}